// MoELayer_80676665688765
// MI455X (gfx1250) — compile-verified
//
#include <hip/hip_runtime.h>
#include <hip/hip_bf16.h>
#include <math.h>

typedef __attribute__((ext_vector_type(2))) float v2f;
typedef __attribute__((ext_vector_type(8))) float v8f;

// D = A(16x4) * B(4x16) + C, fp32, wave32 matrix pipe.
#define WMMA_F32X4(a, b, c) \
  __builtin_amdgcn_wmma_f32_16x16x4_f32(false, (a), false, (b), (short)0, (c), false, false)

static constexpr int Bb = 2, Ss = 2048, Dd = 1024, Ee = 8, Hh = 16, HD = 64;
static constexpr int Tt = Bb * Ss;   // 4096 tokens

#define KP 32           // K panel depth staged in LDS
#define LDS_STRIDE 36   // row stride (floats): 16B-aligned rows, conflict-free b64 frags

// ---- gfx1250 async global->LDS path, guarded so a missing builtin degrades gracefully ----
#if defined(__AMDGCN__) && __has_builtin(__builtin_amdgcn_global_load_async_to_lds_b128) && \
    __has_builtin(__builtin_amdgcn_s_wait_asynccnt)
#define HAVE_ASYNC_LDS 1
#else
#define HAVE_ASYNC_LDS 0
#endif

#if HAVE_ASYNC_LDS
typedef __attribute__((__vector_size__(4 * sizeof(int)))) int v4i;
typedef __attribute__((address_space(1))) v4i* gas_v4i;   // global (AS1) int4*
typedef __attribute__((address_space(3))) v4i* las_v4i;   // LDS    (AS3) int4*
__device__ __forceinline__ void stage16(const float* g, float* l) {
#pragma unroll
  for (int i = 0; i < 4; ++i)
    __builtin_amdgcn_global_load_async_to_lds_b128((gas_v4i)(g + 4 * i), (las_v4i)(l + 4 * i), 0, 0);
}
#else
__device__ __forceinline__ void stage16(const float* g, float* l) {
#pragma unroll
  for (int i = 0; i < 4; ++i)
    *(float4*)(l + 4 * i) = *(const float4*)(g + 4 * i);
}
#endif

// 2x2 subtile WMMA over one K-panel held in LDS
__device__ __forceinline__ void mma_panel(const float* sa, const float* sb,
                                          int wm, int wn, int half, int l16,
                                          v8f& c00, v8f& c01, v8f& c10, v8f& c11) {
#pragma unroll
  for (int kk = 0; kk < KP; kk += 4) {
    int kc = kk + 2 * half;
    v2f a0 = *(const v2f*)(sa + (wm + l16) * LDS_STRIDE + kc);
    v2f a1 = *(const v2f*)(sa + (wm + 16 + l16) * LDS_STRIDE + kc);
    v2f b0 = *(const v2f*)(sb + (wn + l16) * LDS_STRIDE + kc);
    v2f b1 = *(const v2f*)(sb + (wn + 16 + l16) * LDS_STRIDE + kc);
    c00 = WMMA_F32X4(a0, b0, c00);
    c01 = WMMA_F32X4(a0, b1, c01);
    c10 = WMMA_F32X4(a1, b0, c10);
    c11 = WMMA_F32X4(a1, b1, c11);
  }
}

__device__ __forceinline__ void store_tile(float* C, const float* R, v8f c,
                                           int mb, int nb, int N, int half, int l16) {
#pragma unroll
  for (int j = 0; j < 8; ++j) {
    size_t idx = (size_t)(mb + j + 8 * half) * N + nb + l16;
    float v = c[j];
    if (R) v += R[idx];
    C[idx] = v;
  }
}

// ---------------- LayerNorm: one wave per token row ----------------
__global__ void ln_kernel(const float* __restrict__ X, const float* __restrict__ g,
                          const float* __restrict__ bt, float* __restrict__ Y) {
  int lane = threadIdx.x & 31;
  int wave = threadIdx.x >> 5;
  int row  = blockIdx.x * (blockDim.x >> 5) + wave;
  const float* xr = X + (size_t)row * Dd;
  float vals[Dd / 32];
  float s = 0.f;
#pragma unroll
  for (int i = 0; i < Dd / 32; ++i) { vals[i] = xr[lane + i * 32]; s += vals[i]; }
  for (int off = 16; off > 0; off >>= 1) s += __shfl_xor(s, off, 32);
  float mu = s * (1.0f / Dd);
  float vs = 0.f;
#pragma unroll
  for (int i = 0; i < Dd / 32; ++i) { float d = vals[i] - mu; vs += d * d; }
  for (int off = 16; off > 0; off >>= 1) vs += __shfl_xor(vs, off, 32);
  float inv = rsqrtf(vs * (1.0f / Dd) + 1e-5f);
  float* yr = Y + (size_t)row * Dd;
#pragma unroll
  for (int i = 0; i < Dd / 32; ++i) {
    int d = lane + i * 32;
    yr[d] = (vals[i] - mu) * inv * g[d] + bt[d];
  }
}

// ---------------- C[M,N] = A[M,K] * W[N,K]^T (+R) : 64x64 block tile, 4 waves ----------------
__global__ __launch_bounds__(128)
void gemm64_kernel(const float* __restrict__ A, const float* __restrict__ W,
                   const float* __restrict__ R, float* __restrict__ C, int Kd, int N) {
  __shared__ float sa[2][64 * LDS_STRIDE];
  __shared__ float sb[2][64 * LDS_STRIDE];
  const int tid  = threadIdx.x;
  const int lane = tid & 31;
  const int half = lane >> 4, l16 = lane & 15;
  const int wave = tid >> 5;
  const int wm = (wave & 1) * 32, wn = (wave >> 1) * 32;
  const int m0 = blockIdx.y * 64, n0 = blockIdx.x * 64;

  // loader: thread -> 16 consecutive floats of one row
  const int lrow = tid >> 1;
  const int lcol = (tid & 1) * 16;
  const float* ga = A + (size_t)(m0 + lrow) * Kd + lcol;
  const float* gb = W + (size_t)(n0 + lrow) * Kd + lcol;
  const int lofs = lrow * LDS_STRIDE + lcol;

  v8f c00 = {}, c01 = {}, c10 = {}, c11 = {};
  const int npanel = Kd / KP;

#if HAVE_ASYNC_LDS
  stage16(ga, &sa[0][lofs]);
  stage16(gb, &sb[0][lofs]);
  for (int p = 0; p < npanel; ++p) {
    __builtin_amdgcn_s_wait_asynccnt(0);
    __syncthreads();
    if (p + 1 < npanel) {
      stage16(ga + (p + 1) * KP, &sa[(p + 1) & 1][lofs]);
      stage16(gb + (p + 1) * KP, &sb[(p + 1) & 1][lofs]);
    }
    if (p + 2 < npanel) {
      __builtin_prefetch(ga + (p + 2) * KP, 0, 1);
      __builtin_prefetch(gb + (p + 2) * KP, 0, 1);
    }
    mma_panel(sa[p & 1], sb[p & 1], wm, wn, half, l16, c00, c01, c10, c11);
  }
#else
  for (int p = 0; p < npanel; ++p) {
    stage16(ga + p * KP, &sa[p & 1][lofs]);
    stage16(gb + p * KP, &sb[p & 1][lofs]);
    __syncthreads();
    if (p + 1 < npanel) {
      __builtin_prefetch(ga + (p + 1) * KP, 0, 1);
      __builtin_prefetch(gb + (p + 1) * KP, 0, 1);
    }
    mma_panel(sa[p & 1], sb[p & 1], wm, wn, half, l16, c00, c01, c10, c11);
  }
#endif

  store_tile(C, R, c00, m0 + wm,      n0 + wn,      N, half, l16);
  store_tile(C, R, c01, m0 + wm,      n0 + wn + 16, N, half, l16);
  store_tile(C, R, c10, m0 + wm + 16, n0 + wn,      N, half, l16);
  store_tile(C, R, c11, m0 + wm + 16, n0 + wn + 16, N, half, l16);
}

// ---------------- Flash attention: wave per (b, h, 16-query tile) ----------------
__global__ void attn_kernel(const float* __restrict__ Q, const float* __restrict__ Kb,
                            const float* __restrict__ V, float* __restrict__ O) {
  int lane = threadIdx.x & 31;
  int half = lane >> 4, l16 = lane & 15;
  int qt = blockIdx.x, hh = blockIdx.y, bb = blockIdx.z;
  int hoff = hh * HD;
  size_t base = (size_t)bb * Ss;
  int qi0 = qt * 16;

  __shared__ float pt[16][17];
  float mstate[8], lstate[8];
#pragma unroll
  for (int j = 0; j < 8; ++j) { mstate[j] = -1e30f; lstate[j] = 0.f; }
  v8f o0 = {}, o1 = {}, o2 = {}, o3 = {};

  const float* qrow = Q + (base + qi0 + l16) * (size_t)Dd + hoff + 2 * half;

  for (int jt = 0; jt <= qt; ++jt) {
    int ki0 = jt * 16;
    const float* krow = Kb + (base + ki0 + l16) * (size_t)Dd + hoff + 2 * half;
    v8f s = {};
#pragma unroll
    for (int d0 = 0; d0 < HD; d0 += 4) {
      v2f a = *(const v2f*)(qrow + d0);
      v2f b = *(const v2f*)(krow + d0);
      s = WMMA_F32X4(a, b, s);   // S = Q * K^T
    }
    bool diag = (jt == qt);
#pragma unroll
    for (int j = 0; j < 8; ++j) {
      int row = j + 8 * half;
      float val = s[j] * 0.125f;
      if (diag && (ki0 + l16 > qi0 + row)) val = -1e30f;
      float rmax = val;
      for (int off = 8; off > 0; off >>= 1) rmax = fmaxf(rmax, __shfl_xor(rmax, off, 16));
      float mnew = fmaxf(mstate[j], rmax);
      float p    = __expf(val - mnew);
      float corr = __expf(mstate[j] - mnew);
      float psum = p;
      for (int off = 8; off > 0; off >>= 1) psum += __shfl_xor(psum, off, 16);
      lstate[j] = lstate[j] * corr + psum;
      mstate[j] = mnew;
      o0[j] *= corr; o1[j] *= corr; o2[j] *= corr; o3[j] *= corr;
      pt[row][l16] = p;
    }
    __syncthreads();

    const float* vbase = V + (base + ki0) * (size_t)Dd + hoff;
#pragma unroll
    for (int kk = 0; kk < 16; kk += 4) {
      int kr = kk + 2 * half;
      v2f a; a.x = pt[l16][kr]; a.y = pt[l16][kr + 1];
      const float* vr0 = vbase + (size_t)kr * Dd;
      const float* vr1 = vr0 + Dd;
      v2f b;
      b.x = vr0[ 0 + l16]; b.y = vr1[ 0 + l16]; o0 = WMMA_F32X4(a, b, o0);
      b.x = vr0[16 + l16]; b.y = vr1[16 + l16]; o1 = WMMA_F32X4(a, b, o1);
      b.x = vr0[32 + l16]; b.y = vr1[32 + l16]; o2 = WMMA_F32X4(a, b, o2);
      b.x = vr0[48 + l16]; b.y = vr1[48 + l16]; o3 = WMMA_F32X4(a, b, o3);
    }
    __syncthreads();
  }

  float* ob = O + (base + qi0) * (size_t)Dd + hoff;
#pragma unroll
  for (int j = 0; j < 8; ++j) {
    size_t r = (size_t)(j + 8 * half) * Dd;
    float invl = 1.0f / lstate[j];
    ob[r +  0 + l16] = o0[j] * invl;
    ob[r + 16 + l16] = o1[j] * invl;
    ob[r + 32 + l16] = o2[j] * invl;
    ob[r + 48 + l16] = o3[j] * invl;
  }
}

// ---------------- Router + top-2 + softmax gates: wave per token ----------------
__global__ void router_kernel(const float* __restrict__ H2, const float* __restrict__ RW,
                              float* __restrict__ Comb) {
  int lane = threadIdx.x & 31;
  int wave = threadIdx.x >> 5;
  int t = blockIdx.x * (blockDim.x >> 5) + wave;
  const float* hr = H2 + (size_t)t * Dd;
  float logits[Ee];
#pragma unroll
  for (int e = 0; e < Ee; ++e) {
    const float* wr = RW + (size_t)e * Dd;
    float s = 0.f;
    for (int d = lane; d < Dd; d += 32) s += hr[d] * wr[d];
    for (int off = 16; off > 0; off >>= 1) s += __shfl_xor(s, off, 32);
    logits[e] = s;
  }
  int i1 = 0; float v1 = logits[0];
#pragma unroll
  for (int e = 1; e < Ee; ++e) if (logits[e] > v1) { v1 = logits[e]; i1 = e; }
  int i2 = -1; float v2 = -3.0e38f;
#pragma unroll
  for (int e = 0; e < Ee; ++e) if (e != i1 && logits[e] > v2) { v2 = logits[e]; i2 = e; }
  float e2 = __expf(v2 - v1);
  float g1 = 1.0f / (1.0f + e2);
  float g2 = e2 * g1;
  if (lane < Ee) Comb[(size_t)t * Ee + lane] = (lane == i1) ? g1 : ((lane == i2) ? g2 : 0.f);
}

// ---------------- MoE: Out += sum_e gate[t,e] * (H2 * W_e^T), 64x64 tiles, skip inactive ----------------
__global__ __launch_bounds__(128)
void moe64_kernel(const float* __restrict__ H2, const float* __restrict__ EW,
                  const float* __restrict__ Comb, float* __restrict__ Out) {
  __shared__ float sa[2][64 * LDS_STRIDE];
  __shared__ float sb[2][64 * LDS_STRIDE];
  const int tid  = threadIdx.x;
  const int lane = tid & 31;
  const int half = lane >> 4, l16 = lane & 15;
  const int wave = tid >> 5;
  const int wm = (wave & 1) * 32, wn = (wave >> 1) * 32;
  const int m0 = blockIdx.y * 64, n0 = blockIdx.x * 64;

  const int lrow = tid >> 1;
  const int lcol = (tid & 1) * 16;
  const float* ga = H2 + (size_t)(m0 + lrow) * Dd + lcol;
  const int lofs = lrow * LDS_STRIDE + lcol;

  v8f t00 = {}, t01 = {}, t10 = {}, t11 = {};
  const int npanel = Dd / KP;

  for (int e = 0; e < Ee; ++e) {
    float gchk = (tid < 64) ? Comb[(size_t)(m0 + tid) * Ee + e] : 0.f;
    if (!__syncthreads_or(gchk != 0.f)) continue;   // no token in tile routed here
    const float* gb = EW + (size_t)e * Dd * Dd + (size_t)(n0 + lrow) * Dd + lcol;
    v8f c00 = {}, c01 = {}, c10 = {}, c11 = {};

#if HAVE_ASYNC_LDS
    stage16(ga, &sa[0][lofs]);
    stage16(gb, &sb[0][lofs]);
    for (int p = 0; p < npanel; ++p) {
      __builtin_amdgcn_s_wait_asynccnt(0);
      __syncthreads();
      if (p + 1 < npanel) {
        stage16(ga + (p + 1) * KP, &sa[(p + 1) & 1][lofs]);
        stage16(gb + (p + 1) * KP, &sb[(p + 1) & 1][lofs]);
      }
      if (p + 2 < npanel) {
        __builtin_prefetch(gb + (p + 2) * KP, 0, 1);
      }
      mma_panel(sa[p & 1], sb[p & 1], wm, wn, half, l16, c00, c01, c10, c11);
    }
#else
    for (int p = 0; p < npanel; ++p) {
      stage16(ga + p * KP, &sa[p & 1][lofs]);
      stage16(gb + p * KP, &sb[p & 1][lofs]);
      __syncthreads();
      mma_panel(sa[p & 1], sb[p & 1], wm, wn, half, l16, c00, c01, c10, c11);
    }
#endif

#pragma unroll
    for (int j = 0; j < 8; ++j) {
      float g0 = Comb[(size_t)(m0 + wm +      j + 8 * half) * Ee + e];
      float g1 = Comb[(size_t)(m0 + wm + 16 + j + 8 * half) * Ee + e];
      t00[j] += g0 * c00[j]; t01[j] += g0 * c01[j];
      t10[j] += g1 * c10[j]; t11[j] += g1 * c11[j];
    }
  }

#pragma unroll
  for (int j = 0; j < 8; ++j) {
    size_t r0 = (size_t)(m0 + wm +      j + 8 * half) * Dd;
    size_t r1 = (size_t)(m0 + wm + 16 + j + 8 * half) * Dd;
    Out[r0 + n0 + wn      + l16] += t00[j];
    Out[r0 + n0 + wn + 16 + l16] += t01[j];
    Out[r1 + n0 + wn      + l16] += t10[j];
    Out[r1 + n0 + wn + 16 + l16] += t11[j];
  }
}

extern "C" void kernel_launch(void* const* d_in, const int* in_sizes, int n_in,
                              void* d_out, int out_size, void* d_ws, size_t ws_size,
                              hipStream_t stream) {
  (void)in_sizes; (void)n_in; (void)out_size; (void)ws_size;
  const float* x  = (const float*)d_in[0];
  const float* wq = (const float*)d_in[1];
  const float* wk = (const float*)d_in[2];
  const float* wv = (const float*)d_in[3];
  const float* wo = (const float*)d_in[4];
  const float* g1 = (const float*)d_in[5];
  const float* b1 = (const float*)d_in[6];
  const float* g2 = (const float*)d_in[7];
  const float* b2 = (const float*)d_in[8];
  const float* rw = (const float*)d_in[9];
  const float* ew = (const float*)d_in[10];
  float* out = (float*)d_out;
  float* ws  = (float*)d_ws;

  const size_t TD = (size_t)Tt * Dd;
  float* h    = ws;            // LN1 out; reused as attention output
  float* q    = ws + TD;       // reused as h2 after attention
  float* k    = ws + 2 * TD;   // reused as combine after attention
  float* v    = ws + 3 * TD;
  float* attn = h;
  float* h2   = q;
  float* comb = k;

  dim3 lnGrid(Tt / 8), lnBlk(256);
  dim3 gGrid(Dd / 64, Tt / 64), gBlk(128);
  dim3 aGrid(Ss / 16, Hh, Bb), aBlk(32);

  ln_kernel<<<lnGrid, lnBlk, 0, stream>>>(x, g1, b1, h);
  gemm64_kernel<<<gGrid, gBlk, 0, stream>>>(h, wq, nullptr, q, Dd, Dd);
  gemm64_kernel<<<gGrid, gBlk, 0, stream>>>(h, wk, nullptr, k, Dd, Dd);
  gemm64_kernel<<<gGrid, gBlk, 0, stream>>>(h, wv, nullptr, v, Dd, Dd);
  attn_kernel<<<aGrid, aBlk, 0, stream>>>(q, k, v, attn);
  gemm64_kernel<<<gGrid, gBlk, 0, stream>>>(attn, wo, x, out, Dd, Dd);
  ln_kernel<<<lnGrid, lnBlk, 0, stream>>>(out, g2, b2, h2);
  router_kernel<<<lnGrid, lnBlk, 0, stream>>>(h2, rw, comb);
  moe64_kernel<<<gGrid, gBlk, 0, stream>>>(h2, ew, comb, out);
}